// SageLayer1_20529943675141
// MI455X (gfx1250) — compile-verified
//
#include <hip/hip_runtime.h>
#include <hip/hip_bf16.h>

// Problem constants (match reference)
#define B_SZ  4096
#define S_SZ  32
#define F_DIM 256
#define E_DIM 256

typedef __attribute__((ext_vector_type(2))) float v2f;
typedef __attribute__((ext_vector_type(8))) float v8f;
typedef unsigned int u32x4 __attribute__((ext_vector_type(4)));
typedef int i32x4 __attribute__((ext_vector_type(4)));
typedef int i32x8 __attribute__((ext_vector_type(8)));

#if __has_builtin(__builtin_amdgcn_tensor_load_to_lds)
#define USE_TDM 1
#else
#define USE_TDM 0
#endif

__device__ __forceinline__ float wave_reduce_sum(float v) {
#pragma unroll
  for (int off = 16; off >= 1; off >>= 1) v += __shfl_xor(v, off, 32);
  return v;
}
__device__ __forceinline__ float wave_reduce_max(float v) {
#pragma unroll
  for (int off = 16; off >= 1; off >>= 1) v = fmaxf(v, __shfl_xor(v, off, 32));
  return v;
}

// ---------------------------------------------------------------------------
// selfF[b, :] = id2feat[nodes[b], :]   (float4 gather, fully coalesced)
// ---------------------------------------------------------------------------
__global__ __launch_bounds__(256) void gather_self_kernel(
    const int* __restrict__ nodes, const float* __restrict__ id2feat,
    float* __restrict__ selfF) {
  int i  = blockIdx.x * 256 + threadIdx.x;  // over B * (F/4)
  int b  = i >> 6;                          // F/4 == 64
  int f4 = i & 63;
  long row = nodes[b];
  ((float4*)selfF)[i] = ((const float4*)id2feat)[row * 64 + f4];
}

// out[k*256 + n] = in[n*256 + k]  (build Wk^T once; 256 KB, trivial)
__global__ __launch_bounds__(256) void transpose_256_kernel(
    const float* __restrict__ in, float* __restrict__ out) {
  int k = blockIdx.x;
  int n = threadIdx.x;
  out[k * 256 + n] = in[n * 256 + k];
}

#if USE_TDM
// ---------------------------------------------------------------------------
// TDM: DMA a 32x256 fp32 tile (rows k0..k0+31 of W[K,256]) into LDS with a
// 16-byte pad after each 1024-byte row -> LDS row stride of 260 floats.
// Descriptor per CDNA5 ISA ch.8 (group0/group1; groups 2-3 zero for 2-D).
// Issued by one wave; completion via s_wait_tensorcnt + workgroup barrier.
// This toolchain exposes the 6-operand builtin:
//   (u32x4 g0, i32x8 g1, i32x4 g2, i32x4 g3, i32x8 extra, i32 cpol)
// ---------------------------------------------------------------------------
__device__ __forceinline__ void tdm_load_w_tile(const float* W, int k0,
                                                void* lds_dst) {
  unsigned lds_addr = (unsigned)(unsigned long long)lds_dst;  // low 32 bits = LDS offset
  unsigned long long ga =
      (unsigned long long)W + (unsigned long long)k0 * (256ull * 4ull);
  u32x4 g0;
  g0.x = 1u;                                    // count=1, user descriptor
  g0.y = lds_addr;                              // lds_addr [63:32]
  g0.z = (unsigned)(ga & 0xFFFFFFFFu);          // global_addr [95:64]
  g0.w = (unsigned)((ga >> 32) & 0x01FFFFFFu)   // global_addr [120:96]
         | (2u << 30);                          // type=2 ("image") [127:126]
  i32x8 g1 = {
      (int)((2u << 16)      // data_size = 4 bytes
            | (1u << 20)    // pad_enable
            | (7u << 22)    // pad_interval: every 256 DWORDs (1024 B)
            | (3u << 25)),  // pad_amount: 4 DWORDs (16 B)
      (int)(256u << 16),    // tensor_dim0 = 256 (bits 79:48, low half here)
      (int)(0xFFFFu << 16), // tensor_dim1[15:0] = 0xFFFF (huge; no OOB clamp)
      (int)(0x7FFFu | (256u << 16)),  // tensor_dim1[31:16] | tile_dim0 = 256
      32,                   // tile_dim1 = 32 rows, tile_dim2 = 0
      256,                  // tensor_dim0_stride = 256 elements
      0, 0};
  i32x4 gz4 = {0, 0, 0, 0};
  i32x8 gz8 = {0, 0, 0, 0, 0, 0, 0, 0};
  __builtin_amdgcn_tensor_load_to_lds(g0, g1, gz4, gz4, gz8, 0);
}
#endif

// ---------------------------------------------------------------------------
// WMMA GEMM: OUT[16 rows x 256 cols strip] = A[B,K] @ W[K,256] (+bias)
// MODE 0: plain store (row stride ldo).  MODE 1: tanh + L2 row-normalize.
// Block = 128 threads = 4 wave32; wave w owns cols [64w, 64w+64).
// Double-buffered pipeline: TDM DMA of W tile k+1 overlaps WMMA on tile k.
// Fragment layouts per CDNA5 ISA 7.12.2:
//   A 16x4 f32 : lane(m=lane&15, kh=lane>>4), a[j] = A[m][k0 + j + 2*kh]
//   B 4x16 f32 : lane(n=lane&15, kh),         b[j] = W[k0 + j + 2*kh][n]
//   C/D 16x16  : acc[i] -> row = i + 8*kh, col = lane&15
// ---------------------------------------------------------------------------
template <int MODE>
__global__ __launch_bounds__(128) void wmma_gemm_kernel(
    const float* __restrict__ A, int K, const float* __restrict__ W,
    const float* __restrict__ bias, float* __restrict__ out, int ldo) {
  __shared__ float Al[2][16][36];   // stride 36: conflict-free m-indexed reads
  __shared__ float Wl[2][32][260];  // stride 260 == 256 + TDM pad of 4 DWORDs
  __shared__ float rowsq[16];

  const int t    = threadIdx.x;
  const int wave = t >> 5;
  const int lane = t & 31;
  const int kh   = lane >> 4;
  const int nl   = lane & 15;
  const int m    = lane & 15;
  const int b0   = blockIdx.x * 16;
  const int n0w  = wave * 64;

  v8f acc[4];
#pragma unroll
  for (int tt = 0; tt < 4; ++tt)
#pragma unroll
    for (int i = 0; i < 8; ++i) acc[tt][i] = 0.0f;

  const int aRow = t >> 3;        // A tile staging: 128 float4
  const int aC4  = (t & 7) * 4;
  const int nIter = K >> 5;       // K/32 tiles

#if USE_TDM
  // ---- prologue: DMA W tile 0, stage A tile 0 ----
  if (t == 0) tdm_load_w_tile(W, 0, &Wl[0][0][0]);
  {
    float4 v = *(const float4*)(A + (long)(b0 + aRow) * K + aC4);
    Al[0][aRow][aC4 + 0] = v.x; Al[0][aRow][aC4 + 1] = v.y;
    Al[0][aRow][aC4 + 2] = v.z; Al[0][aRow][aC4 + 3] = v.w;
  }
  for (int it = 0; it < nIter; ++it) {
    const int cur = it & 1, nxt = cur ^ 1;
    if (t == 0) __builtin_amdgcn_s_wait_tensorcnt(0);  // W[cur] DMA complete
    __syncthreads();  // publish W[cur] + A[cur]; prior compute done
    if (it + 1 < nIter) {
      if (t == 0) tdm_load_w_tile(W, (it + 1) * 32, &Wl[nxt][0][0]);
      float4 v =
          *(const float4*)(A + (long)(b0 + aRow) * K + (it + 1) * 32 + aC4);
      Al[nxt][aRow][aC4 + 0] = v.x; Al[nxt][aRow][aC4 + 1] = v.y;
      Al[nxt][aRow][aC4 + 2] = v.z; Al[nxt][aRow][aC4 + 3] = v.w;
    }
#pragma unroll
    for (int kk = 0; kk < 32; kk += 4) {
      v2f a;
      a[0] = Al[cur][m][kk + 2 * kh + 0];
      a[1] = Al[cur][m][kk + 2 * kh + 1];
#pragma unroll
      for (int tt = 0; tt < 4; ++tt) {
        const int n = n0w + tt * 16 + nl;
        v2f b;
        b[0] = Wl[cur][kk + 2 * kh + 0][n];
        b[1] = Wl[cur][kk + 2 * kh + 1][n];
        acc[tt] = __builtin_amdgcn_wmma_f32_16x16x4_f32(
            false, a, false, b, (short)0, acc[tt], false, false);
      }
    }
  }
#else
  // ---- fallback: synchronous staging (no TDM builtin available) ----
  for (int it = 0; it < nIter; ++it) {
    const int k0 = it * 32;
    {
      float4 v = *(const float4*)(A + (long)(b0 + aRow) * K + k0 + aC4);
      Al[0][aRow][aC4 + 0] = v.x; Al[0][aRow][aC4 + 1] = v.y;
      Al[0][aRow][aC4 + 2] = v.z; Al[0][aRow][aC4 + 3] = v.w;
    }
#pragma unroll
    for (int li = 0; li < 16; ++li) {
      int i   = li * 128 + t;
      int row = i >> 6;
      int c4  = (i & 63) * 4;
      float4 v = *(const float4*)(W + (long)(k0 + row) * 256 + c4);
      Wl[0][row][c4 + 0] = v.x; Wl[0][row][c4 + 1] = v.y;
      Wl[0][row][c4 + 2] = v.z; Wl[0][row][c4 + 3] = v.w;
    }
    __syncthreads();
#pragma unroll
    for (int kk = 0; kk < 32; kk += 4) {
      v2f a;
      a[0] = Al[0][m][kk + 2 * kh + 0];
      a[1] = Al[0][m][kk + 2 * kh + 1];
#pragma unroll
      for (int tt = 0; tt < 4; ++tt) {
        const int n = n0w + tt * 16 + nl;
        v2f b;
        b[0] = Wl[0][kk + 2 * kh + 0][n];
        b[1] = Wl[0][kk + 2 * kh + 1][n];
        acc[tt] = __builtin_amdgcn_wmma_f32_16x16x4_f32(
            false, a, false, b, (short)0, acc[tt], false, false);
      }
    }
    __syncthreads();
  }
#endif

  // Epilogue: bias (+tanh) in registers
  float vals[4][8];
#pragma unroll
  for (int tt = 0; tt < 4; ++tt) {
    const int col = n0w + tt * 16 + nl;
    const float bb = bias ? bias[col] : 0.0f;
#pragma unroll
    for (int i = 0; i < 8; ++i) {
      float v = acc[tt][i] + bb;
      if (MODE == 1) v = tanhf(v);
      vals[tt][i] = v;
    }
  }

  if (MODE == 0) {
#pragma unroll
    for (int tt = 0; tt < 4; ++tt) {
      const int col = n0w + tt * 16 + nl;
#pragma unroll
      for (int i = 0; i < 8; ++i)
        out[(long)(b0 + i + 8 * kh) * ldo + col] = vals[tt][i];
    }
  } else {
    if (t < 16) rowsq[t] = 0.0f;
    __syncthreads();
#pragma unroll
    for (int i = 0; i < 8; ++i) {
      float s = 0.0f;
#pragma unroll
      for (int tt = 0; tt < 4; ++tt) s += vals[tt][i] * vals[tt][i];
      atomicAdd(&rowsq[i + 8 * kh], s);  // ds_add_f32
    }
    __syncthreads();
#pragma unroll
    for (int i = 0; i < 8; ++i) {
      const int r = i + 8 * kh;
      const float inv = 1.0f / fmaxf(sqrtf(rowsq[r]), 1e-12f);
#pragma unroll
      for (int tt = 0; tt < 4; ++tt) {
        const int col = n0w + tt * 16 + nl;
        out[(long)(b0 + r) * ldo + col] = vals[tt][i] * inv;
      }
    }
  }
}

// ---------------------------------------------------------------------------
// Fused attention + mean aggregation (one block per batch row).
//   score_s = feat[attn_s] . qt_b        (bk.q dropped: softmax shift-invariant)
//   xattn_b = sum_s softmax(score)_s * feat[attn_s]
//   xmean_b = mean_s feat[mean_s]
// 32 neighbor rows staged in LDS (32 KB); wave-parallel dots + shfl softmax.
// ---------------------------------------------------------------------------
__global__ __launch_bounds__(256) void attn_mean_kernel(
    const int* __restrict__ neigh_attn, const int* __restrict__ neigh_mean,
    const float* __restrict__ id2feat, const float* __restrict__ qt,
    float* __restrict__ xattn, float* __restrict__ xmean) {
  __shared__ float xn[S_SZ][F_DIM + 1];
  __shared__ float qts[F_DIM];
  __shared__ float sc[S_SZ];
  __shared__ int ia[S_SZ];
  __shared__ int im[S_SZ];

  const int b = blockIdx.x;
  const int t = threadIdx.x;
  const int wave = t >> 5, lane = t & 31;

  if (t < S_SZ) {
    int ra = neigh_attn[b * S_SZ + t];
    int rm = neigh_mean[b * S_SZ + t];
    ia[t] = ra;
    im[t] = rm;
    __builtin_prefetch(id2feat + (long)rm * F_DIM, 0, 1);  // global_prefetch_b8
  }
  qts[t] = qt[(long)b * E_DIM + t];
  __syncthreads();

#pragma unroll 4
  for (int s = 0; s < S_SZ; ++s)
    xn[s][t] = id2feat[(long)ia[s] * F_DIM + t];
  __syncthreads();

  // scores: wave w handles s = w, w+8, w+16, w+24
  for (int s = wave; s < S_SZ; s += 8) {
    float acc = 0.0f;
#pragma unroll
    for (int i = 0; i < 8; ++i)
      acc += xn[s][lane + 32 * i] * qts[lane + 32 * i];
    acc = wave_reduce_sum(acc);
    if (lane == 0) sc[s] = acc;
  }
  __syncthreads();

  if (wave == 0) {  // softmax over 32 scores in one wave32
    float v  = sc[lane];
    float mx = wave_reduce_max(v);
    float e  = __expf(v - mx);
    float sm = wave_reduce_sum(e);
    sc[lane] = e / sm;
  }
  __syncthreads();

  float msum = 0.0f, asum = 0.0f;
#pragma unroll 4
  for (int s = 0; s < S_SZ; ++s) {
    msum += id2feat[(long)im[s] * F_DIM + t];
    asum += sc[s] * xn[s][t];
  }
  xmean[(long)b * F_DIM + t] = msum * (1.0f / S_SZ);
  xattn[(long)b * F_DIM + t] = asum;
}

// ---------------------------------------------------------------------------
extern "C" void kernel_launch(void* const* d_in, const int* in_sizes, int n_in,
                              void* d_out, int out_size, void* d_ws,
                              size_t ws_size, hipStream_t stream) {
  const int*   nodes      = (const int*)d_in[0];
  const int*   neigh_mean = (const int*)d_in[1];
  const int*   neigh_attn = (const int*)d_in[2];
  const float* id2feat    = (const float*)d_in[3];
  const float* Wm = (const float*)d_in[4];
  const float* bm = (const float*)d_in[5];
  const float* Wq = (const float*)d_in[6];
  const float* bq = (const float*)d_in[7];
  const float* Wk = (const float*)d_in[8];
  // d_in[9] = bk: provably unused (uniform shift under softmax)
  const float* Wv = (const float*)d_in[10];
  const float* bv = (const float*)d_in[11];
  const float* Wc = (const float*)d_in[12];
  const float* bc = (const float*)d_in[13];
  float* out = (float*)d_out;

  const long BF = (long)B_SZ * F_DIM;  // 1,048,576 floats
  float* ws    = (float*)d_ws;
  float* selfF = ws + 0 * BF;
  float* q     = ws + 1 * BF;
  float* qt    = ws + 2 * BF;
  float* xattn = ws + 3 * BF;
  float* xmean = ws + 4 * BF;
  float* comb  = ws + 5 * BF;       // [B, 512] = [mix | ctx]
  float* WkT   = ws + 7 * BF;       // 256x256

  gather_self_kernel<<<B_SZ * (F_DIM / 4) / 256, 256, 0, stream>>>(
      nodes, id2feat, selfF);
  transpose_256_kernel<<<256, 256, 0, stream>>>(Wk, WkT);

  // q = selfF @ Wq + bq
  wmma_gemm_kernel<0><<<B_SZ / 16, 128, 0, stream>>>(selfF, 256, Wq, bq, q, 256);
  // qt = q @ Wk^T   (no bias; bk dropped by softmax shift invariance)
  wmma_gemm_kernel<0><<<B_SZ / 16, 128, 0, stream>>>(q, 256, WkT, nullptr, qt, 256);

  attn_mean_kernel<<<B_SZ, 256, 0, stream>>>(neigh_attn, neigh_mean, id2feat,
                                             qt, xattn, xmean);

  // mix = xattn @ Wv + bv  -> comb[:, 0:256]
  wmma_gemm_kernel<0><<<B_SZ / 16, 128, 0, stream>>>(xattn, 256, Wv, bv, comb, 512);
  // ctx = xmean @ Wm + bm  -> comb[:, 256:512]
  wmma_gemm_kernel<0><<<B_SZ / 16, 128, 0, stream>>>(xmean, 256, Wm, bm, comb + 256, 512);

  // out = l2norm(tanh(comb @ Wc + bc))
  wmma_gemm_kernel<1><<<B_SZ / 16, 128, 0, stream>>>(comb, 512, Wc, bc, out, 256);
}